// MagNet_35450660061787
// MI455X (gfx1250) — compile-verified
//
#include <hip/hip_runtime.h>
#include <hip/hip_bf16.h>
#include <math.h>
#include <stdint.h>

// ---------------------------------------------------------------------------
// MagNet forward for MI455X (gfx1250, wave32, WMMA).
// Matmuls: V_WMMA_F32_16X16X4_F32 (fp32 WMMA path).
// Tile fills: GLOBAL_LOAD_ASYNC_TO_LDS_B128 + ASYNCcnt, double-buffered LDS.
// ---------------------------------------------------------------------------

typedef float v2f __attribute__((ext_vector_type(2)));
typedef float v8f __attribute__((ext_vector_type(8)));

#define B_  64
#define N_  512
#define D_  256
#define E_  4096
#define NN_ (N_ * N_)      // 262144
#define ND_ (N_ * D_)      // 131072

#define LDA_S 36           // gemm_rr A-tile LDS row stride (KT=32, 16B aligned)
#define LDB_S 68           // B-tile LDS row stride (16B aligned)
#define LDA16 20           // cgemm A-tile LDS row stride (KT=16, 16B aligned)

__device__ __forceinline__ v8f wmma4(v2f a, v2f b, v8f c) {
  // D = A(16x4 f32) * B(4x16 f32) + C(16x16 f32)
  return __builtin_amdgcn_wmma_f32_16x16x4_f32(
      false, a, false, b, (short)0, c, false, false);
}

// Async global -> LDS (16B per lane); tracked by ASYNCcnt.
__device__ __forceinline__ void async_b128(uint32_t lds_addr, const void* gptr) {
  asm volatile("global_load_async_to_lds_b128 %0, %1, off"
               :: "v"(lds_addr), "v"((uint64_t)(uintptr_t)gptr)
               : "memory");
}
__device__ __forceinline__ uint32_t lds_off(const void* p) {
  return (uint32_t)(uintptr_t)p;   // LDS aperture: low 32 bits = LDS address
}

// ---------------------------------------------------------------------------
// Graph construction
// ---------------------------------------------------------------------------

__global__ void scatter_edges(const int* __restrict__ senders,
                              const int* __restrict__ receivers,
                              float* __restrict__ adj) {
  int idx = blockIdx.x * 256 + threadIdx.x;       // b*E_ threads
  if (idx >= B_ * E_) return;
  int b = idx >> 12;                              // E_ = 4096
  int s = senders[idx];
  int r = receivers[idx];
  if (s >= 0 && s < N_ && r >= 0 && r < N_) {
    adj[(size_t)b * NN_ + (size_t)s * N_ + r] = 1.0f;  // clamp(adj,1) == 1
  }
}

__global__ void build_deg(const float* __restrict__ adj,
                          float* __restrict__ deg) {
  int idx = blockIdx.x * 256 + threadIdx.x;       // b*N_ threads
  if (idx >= B_ * N_) return;
  int b = idx >> 9;
  int j = idx & (N_ - 1);
  const float* base = adj + (size_t)b * NN_;
  const float4* row4 = (const float4*)(base + (size_t)j * N_);
  float sum = 0.f;
  for (int i4 = 0; i4 < N_ / 4; ++i4) {
    float4 a4 = row4[i4];
#pragma unroll
    for (int k = 0; k < 4; ++k) {
      float a = (k == 0) ? a4.x : (k == 1) ? a4.y : (k == 2) ? a4.z : a4.w;
      float at = base[(size_t)(i4 * 4 + k) * N_ + j];
      float s = a + at;
      if (a != 0.f && at != 0.f) s *= 0.5f;
      sum += s;
    }
  }
  deg[idx] = sum;
}

__global__ void build_lap(const float* __restrict__ adj,
                          const float* __restrict__ deg,
                          const int* __restrict__ n_node,
                          float* __restrict__ lapRe,
                          float* __restrict__ lapIm) {
  int idx = blockIdx.x * 256 + threadIdx.x;       // b*NN_ threads
  if (idx >= B_ * NN_) return;
  int b = idx >> 18;
  int rem = idx & (NN_ - 1);
  int i = rem >> 9;
  int j = rem & (N_ - 1);
  const float* base = adj + (size_t)b * NN_;
  float a  = base[(size_t)i * N_ + j];
  float at = base[(size_t)j * N_ + i];
  float sym = a + at;
  if (a != 0.f && at != 0.f) sym *= 0.5f;
  float di = deg[b * N_ + i];
  float dj = deg[b * N_ + j];
  float nrm = rsqrtf(fmaxf(di, 1.f)) * sym * rsqrtf(fmaxf(dj, 1.f));
  // phase = (pi/2)*(a-at); a,at in {0,1} -> cos in {0,1}, sin in {-1,0,1} exact
  float d = a - at;
  float cosd = (d == 0.f) ? 1.f : 0.f;
  float sind = (d > 0.f) ? 1.f : ((d < 0.f) ? -1.f : 0.f);
  float lre = ((i == j) ? 1.f : 0.f) - nrm * cosd;
  float lim = -nrm * sind;
  int nn = n_node[b];
  float mask = (i < nn && j < nn) ? 1.f : 0.f;
  lapRe[idx] = mask * lre;
  lapIm[idx] = mask * lim;
}

// ---------------------------------------------------------------------------
// Power iteration for lambda_max (Hermitian PSD -> converges to max eig)
// ---------------------------------------------------------------------------

__global__ void init_power(float* __restrict__ vRe, float* __restrict__ vIm) {
  int idx = blockIdx.x * 256 + threadIdx.x;
  if (idx >= B_ * N_) return;
  int b = idx >> 9;
  int j = idx & (N_ - 1);
  vRe[idx] = 1.0f + 0.25f * __sinf(0.37f * (float)j + 0.11f * (float)b);
  vIm[idx] = 0.25f * __cosf(0.53f * (float)j);
}

__global__ __launch_bounds__(256) void power_iter(
    const float* __restrict__ lapRe, const float* __restrict__ lapIm,
    float* __restrict__ vRe, float* __restrict__ vIm,
    float* __restrict__ lam) {
  __shared__ float sre[N_];
  __shared__ float sim[N_];
  __shared__ float red[256];
  int b = blockIdx.x;
  int t = threadIdx.x;
  const float* Lre = lapRe + (size_t)b * NN_;
  const float* Lim = lapIm + (size_t)b * NN_;
  sre[t]       = vRe[b * N_ + t];
  sre[t + 256] = vRe[b * N_ + t + 256];
  sim[t]       = vIm[b * N_ + t];
  sim[t + 256] = vIm[b * N_ + t + 256];
  __syncthreads();
  float wre[2], wim[2];
  for (int rr = 0; rr < 2; ++rr) {
    int j = t + rr * 256;
    const float4* rowRe = (const float4*)(Lre + (size_t)j * N_);
    const float4* rowIm = (const float4*)(Lim + (size_t)j * N_);
    float are = 0.f, aim = 0.f;
    for (int i4 = 0; i4 < N_ / 4; ++i4) {
      float4 lr4 = rowRe[i4];
      float4 li4 = rowIm[i4];
#pragma unroll
      for (int k = 0; k < 4; ++k) {
        float lr = (k == 0) ? lr4.x : (k == 1) ? lr4.y : (k == 2) ? lr4.z : lr4.w;
        float li = (k == 0) ? li4.x : (k == 1) ? li4.y : (k == 2) ? li4.z : li4.w;
        float xr = sre[i4 * 4 + k];
        float xi = sim[i4 * 4 + k];
        are += lr * xr - li * xi;
        aim += lr * xi + li * xr;
      }
    }
    wre[rr] = are;
    wim[rr] = aim;
  }
  red[t] = wre[0]*wre[0] + wim[0]*wim[0] + wre[1]*wre[1] + wim[1]*wim[1];
  __syncthreads();
  for (int s = 128; s > 0; s >>= 1) {
    if (t < s) red[t] += red[t + s];
    __syncthreads();
  }
  float nrm = sqrtf(red[0]);
  float inv = (nrm > 0.f) ? 1.f / nrm : 0.f;
  for (int rr = 0; rr < 2; ++rr) {
    vRe[b * N_ + t + rr * 256] = wre[rr] * inv;
    vIm[b * N_ + t + rr * 256] = wim[rr] * inv;
  }
  if (t == 0) lam[b] = nrm;
}

__global__ void finalize_scale(const float* __restrict__ lam,
                               float* __restrict__ scl) {
  __shared__ float red[64];
  int t = threadIdx.x;
  red[t] = lam[t];
  __syncthreads();
  for (int s = 32; s > 0; s >>= 1) {
    if (t < s) red[t] = fmaxf(red[t], red[t + s]);
    __syncthreads();
  }
  if (t == 0) scl[0] = 2.0f / fmaxf(red[0], 1e-30f);
}

__global__ void make_t1(float* __restrict__ lapRe, float* __restrict__ lapIm,
                        const float* __restrict__ scl) {
  int idx4 = blockIdx.x * 256 + threadIdx.x;      // b*NN_/4 threads
  if (idx4 >= B_ * NN_ / 4) return;
  int rem4 = idx4 & (NN_ / 4 - 1);
  int i = rem4 >> 7;                              // 128 float4 per row
  int j0 = (rem4 & 127) * 4;
  float s = scl[0];
  float4* pr = (float4*)lapRe + idx4;
  float4* pi = (float4*)lapIm + idx4;
  float4 re = *pr, im = *pi;
  re.x = s * re.x - ((i == j0 + 0) ? 1.f : 0.f);
  re.y = s * re.y - ((i == j0 + 1) ? 1.f : 0.f);
  re.z = s * re.z - ((i == j0 + 2) ? 1.f : 0.f);
  re.w = s * re.w - ((i == j0 + 3) ? 1.f : 0.f);
  im.x *= s; im.y *= s; im.z *= s; im.w *= s;
  *pr = re;
  *pi = im;
}

// ---------------------------------------------------------------------------
// Real WMMA GEMM: Out[M,Nc] = A[M,K] @ W[K,Nc] (+bias) (+Cin)
// Block tile 128x64, KT=32, double-buffered async-LDS fills.
// Wave = one 16-row strip x four 16-col strips.
// ---------------------------------------------------------------------------

__global__ __launch_bounds__(256) void gemm_rr(
    const float* __restrict__ A, const float* __restrict__ W,
    const float* __restrict__ bias, const float* __restrict__ Cin,
    float* __restrict__ Out, int K, int Nc) {
  __shared__ __align__(16) float As[2][128 * LDA_S];
  __shared__ __align__(16) float Bs[2][32 * LDB_S];
  int tid = threadIdx.x;
  int mBase = blockIdx.y * 128;
  int nBase = blockIdx.x * 64;
  int lane = tid & 31, wave = tid >> 5;
  int half = lane >> 4, l = lane & 15;
  int mW = wave * 16;
  v8f acc[4];
#pragma unroll
  for (int s = 0; s < 4; ++s) acc[s] = {0.f,0.f,0.f,0.f,0.f,0.f,0.f,0.f};

  // issue one chunk's tiles (6 async instructions per wave)
  auto issue = [&](int kc, int bsel) {
#pragma unroll
    for (int i = 0; i < 4; ++i) {
      int f = tid + i * 256;                 // A tile 128x32 (1024 float4)
      int r = f >> 3, c4 = f & 7;
      async_b128(lds_off(&As[bsel][r * LDA_S + c4 * 4]),
                 A + (size_t)(mBase + r) * K + kc + c4 * 4);
    }
#pragma unroll
    for (int i = 0; i < 2; ++i) {
      int f = tid + i * 256;                 // B tile 32x64 (512 float4)
      int r = f >> 4, c4 = f & 15;
      async_b128(lds_off(&Bs[bsel][r * LDB_S + c4 * 4]),
                 W + (size_t)(kc + r) * Nc + nBase + c4 * 4);
    }
  };

  issue(0, 0);
  int nch = K / 32;
  for (int c = 0; c < nch; ++c) {
    int cur = c & 1;
    if (c + 1 < nch) {
      issue((c + 1) * 32, cur ^ 1);
      asm volatile("s_wait_asynccnt 0x6" ::: "memory");  // chunk c resident
    } else {
      asm volatile("s_wait_asynccnt 0x0" ::: "memory");
    }
    __syncthreads();
    const float* Asb = As[cur];
    const float* Bsb = Bs[cur];
#pragma unroll
    for (int k0 = 0; k0 < 32; k0 += 4) {
      int ak = k0 + 2 * half;
      v2f a;
      a.x = Asb[(mW + l) * LDA_S + ak];
      a.y = Asb[(mW + l) * LDA_S + ak + 1];
#pragma unroll
      for (int s = 0; s < 4; ++s) {
        v2f b;
        b.x = Bsb[ak * LDB_S + s * 16 + l];
        b.y = Bsb[(ak + 1) * LDB_S + s * 16 + l];
        acc[s] = wmma4(a, b, acc[s]);
      }
    }
    __syncthreads();   // buffer reusable for chunk c+2
  }
#pragma unroll
  for (int s = 0; s < 4; ++s) {
    int col = nBase + s * 16 + l;
    float bv = bias ? bias[col] : 0.f;
#pragma unroll
    for (int v = 0; v < 8; ++v) {
      int row = mBase + mW + 8 * half + v;
      size_t o = (size_t)row * Nc + col;
      Out[o] = acc[s][v] + bv + (Cin ? Cin[o] : 0.f);
    }
  }
}

// ---------------------------------------------------------------------------
// Batched complex WMMA GEMM: Z = T1 @ X  (T1: [512,512] cplx, X: [512,256] cplx)
// Zre = Tre*Xre - Tim*Xim ; Zim = Tre*Xim + Tim*Xre
// Block tile 128x64, KT=16, double-buffered async-LDS fills.
// ---------------------------------------------------------------------------

__global__ __launch_bounds__(256) void cgemm_t1x(
    const float* __restrict__ Tre, const float* __restrict__ Tim,
    const float* __restrict__ Xre, const float* __restrict__ Xim,
    float* __restrict__ Zre, float* __restrict__ Zim) {
  __shared__ __align__(16) float Ars[2][128 * LDA16];
  __shared__ __align__(16) float Ais[2][128 * LDA16];
  __shared__ __align__(16) float Brs[2][16 * LDB_S];
  __shared__ __align__(16) float Bis[2][16 * LDB_S];
  int b = blockIdx.z;
  const float* Ar = Tre + (size_t)b * NN_;
  const float* Ai = Tim + (size_t)b * NN_;
  const float* Br = Xre + (size_t)b * ND_;
  const float* Bi = Xim + (size_t)b * ND_;
  float* Or = Zre + (size_t)b * ND_;
  float* Oi = Zim + (size_t)b * ND_;
  int tid = threadIdx.x;
  int mBase = blockIdx.y * 128;
  int nBase = blockIdx.x * 64;
  int lane = tid & 31, wave = tid >> 5;
  int half = lane >> 4, l = lane & 15;
  int mW = wave * 16;
  v8f accr[4], acci[4];
#pragma unroll
  for (int s = 0; s < 4; ++s) {
    accr[s] = {0.f,0.f,0.f,0.f,0.f,0.f,0.f,0.f};
    acci[s] = {0.f,0.f,0.f,0.f,0.f,0.f,0.f,0.f};
  }

  // issue one KT=16 chunk: A planes 128x16, B planes 16x64 (6 async per wave)
  auto issue = [&](int kc, int bsel) {
#pragma unroll
    for (int i = 0; i < 2; ++i) {
      int f = tid + i * 256;                 // 512 float4 per A plane
      int r = f >> 2, c4 = f & 3;
      size_t go = (size_t)(mBase + r) * N_ + kc + c4 * 4;
      async_b128(lds_off(&Ars[bsel][r * LDA16 + c4 * 4]), Ar + go);
      async_b128(lds_off(&Ais[bsel][r * LDA16 + c4 * 4]), Ai + go);
    }
    {
      int r = tid >> 4, c4 = tid & 15;       // 256 float4 per B plane
      size_t go = (size_t)(kc + r) * D_ + nBase + c4 * 4;
      async_b128(lds_off(&Brs[bsel][r * LDB_S + c4 * 4]), Br + go);
      async_b128(lds_off(&Bis[bsel][r * LDB_S + c4 * 4]), Bi + go);
    }
  };

  issue(0, 0);
  const int nch = N_ / 16;                   // 32 chunks
  for (int c = 0; c < nch; ++c) {
    int cur = c & 1;
    if (c + 1 < nch) {
      issue((c + 1) * 16, cur ^ 1);
      asm volatile("s_wait_asynccnt 0x6" ::: "memory");
    } else {
      asm volatile("s_wait_asynccnt 0x0" ::: "memory");
    }
    __syncthreads();
    const float* Arb = Ars[cur];
    const float* Aib = Ais[cur];
    const float* Brb = Brs[cur];
    const float* Bib = Bis[cur];
#pragma unroll
    for (int k0 = 0; k0 < 16; k0 += 4) {
      int ak = k0 + 2 * half;
      v2f ar, ai, nai;
      ar.x = Arb[(mW + l) * LDA16 + ak];
      ar.y = Arb[(mW + l) * LDA16 + ak + 1];
      ai.x = Aib[(mW + l) * LDA16 + ak];
      ai.y = Aib[(mW + l) * LDA16 + ak + 1];
      nai.x = -ai.x;
      nai.y = -ai.y;
#pragma unroll
      for (int s = 0; s < 4; ++s) {
        v2f br, bi;
        br.x = Brb[ak * LDB_S + s * 16 + l];
        br.y = Brb[(ak + 1) * LDB_S + s * 16 + l];
        bi.x = Bib[ak * LDB_S + s * 16 + l];
        bi.y = Bib[(ak + 1) * LDB_S + s * 16 + l];
        accr[s] = wmma4(ar, br, accr[s]);
        accr[s] = wmma4(nai, bi, accr[s]);
        acci[s] = wmma4(ar, bi, acci[s]);
        acci[s] = wmma4(ai, br, acci[s]);
      }
    }
    __syncthreads();
  }
#pragma unroll
  for (int s = 0; s < 4; ++s) {
    int col = nBase + s * 16 + l;
#pragma unroll
    for (int v = 0; v < 8; ++v) {
      int row = mBase + mW + 8 * half + v;
      Or[(size_t)row * D_ + col] = accr[s][v];
      Oi[(size_t)row * D_ + col] = acci[s][v];
    }
  }
}

// ---------------------------------------------------------------------------
// complex_relu: mask = (re >= 0) applied to both planes; configurable out ld
// ---------------------------------------------------------------------------

__global__ void relu_combine(const float* __restrict__ Yre,
                             const float* __restrict__ Yim,
                             float* __restrict__ outRe,
                             float* __restrict__ outIm, int ld) {
  int idx4 = blockIdx.x * 256 + threadIdx.x;      // B_*ND_/4 threads
  if (idx4 >= B_ * ND_ / 4) return;
  int r = idx4 >> 6;                              // 64 float4 per 256-col row
  int c = (idx4 & 63) * 4;
  float4 yr = ((const float4*)Yre)[idx4];
  float4 yi = ((const float4*)Yim)[idx4];
  float4 orv, oiv;
  float m;
  m = (yr.x >= 0.f) ? 1.f : 0.f; orv.x = m * yr.x; oiv.x = m * yi.x;
  m = (yr.y >= 0.f) ? 1.f : 0.f; orv.y = m * yr.y; oiv.y = m * yi.y;
  m = (yr.z >= 0.f) ? 1.f : 0.f; orv.z = m * yr.z; oiv.z = m * yi.z;
  m = (yr.w >= 0.f) ? 1.f : 0.f; orv.w = m * yr.w; oiv.w = m * yi.w;
  *(float4*)(outRe + (size_t)r * ld + c) = orv;
  *(float4*)(outIm + (size_t)r * ld + c) = oiv;
}

// ---------------------------------------------------------------------------
// Host-side launch
// ---------------------------------------------------------------------------

extern "C" void kernel_launch(void* const* d_in, const int* in_sizes, int n_in,
                              void* d_out, int out_size, void* d_ws, size_t ws_size,
                              hipStream_t stream) {
  (void)in_sizes; (void)n_in; (void)out_size; (void)ws_size;
  const float* nodes     = (const float*)d_in[0];
  const int*   senders   = (const int*)d_in[1];
  const int*   receivers = (const int*)d_in[2];
  const int*   n_node    = (const int*)d_in[3];
  const float* w0_0      = (const float*)d_in[4];
  const float* b0_0      = (const float*)d_in[5];
  const float* w1_0      = (const float*)d_in[6];
  const float* w0_1      = (const float*)d_in[7];
  const float* b0_1      = (const float*)d_in[8];
  const float* w1_1      = (const float*)d_in[9];
  float* out = (float*)d_out;
  float* ws  = (float*)d_ws;

  // Workspace layout (floats)
  float* lapRe = ws;
  float* lapIm = lapRe + (size_t)B_ * NN_;
  float* adj   = lapIm + (size_t)B_ * NN_;     // reused as Z after build_lap
  float* Zre   = adj;
  float* Zim   = adj + (size_t)B_ * ND_;
  float* Xre   = adj + (size_t)B_ * NN_;
  float* Xim   = Xre + (size_t)B_ * ND_;
  float* Yre   = Xim + (size_t)B_ * ND_;
  float* Yim   = Yre + (size_t)B_ * ND_;
  float* deg   = Yim + (size_t)B_ * ND_;
  float* vRe   = deg + B_ * N_;
  float* vIm   = vRe + B_ * N_;
  float* lam   = vIm + B_ * N_;
  float* scl   = lam + B_;

  // --- Laplacian construction ---
  hipMemsetAsync(adj, 0, (size_t)B_ * NN_ * sizeof(float), stream);
  scatter_edges<<<(B_ * E_) / 256, 256, 0, stream>>>(senders, receivers, adj);
  build_deg<<<(B_ * N_) / 256, 256, 0, stream>>>(adj, deg);
  build_lap<<<(B_ * NN_) / 256, 256, 0, stream>>>(adj, deg, n_node, lapRe, lapIm);

  // --- lambda_max via power iteration; t1 = (2/lmax)*lap - I ---
  init_power<<<(B_ * N_) / 256, 256, 0, stream>>>(vRe, vIm);
  for (int it = 0; it < 50; ++it)
    power_iter<<<B_, 256, 0, stream>>>(lapRe, lapIm, vRe, vIm, lam);
  finalize_scale<<<1, 64, 0, stream>>>(lam, scl);
  make_t1<<<(B_ * NN_ / 4) / 256, 256, 0, stream>>>(lapRe, lapIm, scl);

  hipMemsetAsync(Xim, 0, (size_t)B_ * ND_ * sizeof(float), stream);

  dim3 gG(D_ / 64, (B_ * N_) / 128);  // (4, 256) real GEMM over flattened rows
  dim3 gC(D_ / 64, N_ / 128, B_);     // (4, 4, 64) batched complex GEMM

  // --- layer 1 (Xre = nodes, Xim = 0) ---
  gemm_rr<<<gG, 256, 0, stream>>>(nodes, w0_0, b0_0, nullptr, Yre, D_, D_);
  gemm_rr<<<gG, 256, 0, stream>>>(Xim,   w0_0, b0_0, nullptr, Yim, D_, D_);
  cgemm_t1x<<<gC, 256, 0, stream>>>(lapRe, lapIm, nodes, Xim, Zre, Zim);
  gemm_rr<<<gG, 256, 0, stream>>>(Zre, w1_0, nullptr, Yre, Yre, D_, D_);
  gemm_rr<<<gG, 256, 0, stream>>>(Zim, w1_0, nullptr, Yim, Yim, D_, D_);
  relu_combine<<<(B_ * ND_ / 4) / 256, 256, 0, stream>>>(Yre, Yim, Xre, Xim, D_);

  // --- layer 2 ---
  gemm_rr<<<gG, 256, 0, stream>>>(Xre, w0_1, b0_1, nullptr, Yre, D_, D_);
  gemm_rr<<<gG, 256, 0, stream>>>(Xim, w0_1, b0_1, nullptr, Yim, D_, D_);
  cgemm_t1x<<<gC, 256, 0, stream>>>(lapRe, lapIm, Xre, Xim, Zre, Zim);
  gemm_rr<<<gG, 256, 0, stream>>>(Zre, w1_1, nullptr, Yre, Yre, D_, D_);
  gemm_rr<<<gG, 256, 0, stream>>>(Zim, w1_1, nullptr, Yim, Yim, D_, D_);
  // final: write concat(re, im) layout [b, n, 512] directly
  relu_combine<<<(B_ * ND_ / 4) / 256, 256, 0, stream>>>(Yre, Yim, out, out + D_, 2 * D_);
}